// Upsampling_7705171329302
// MI455X (gfx1250) — compile-verified
//
#include <hip/hip_runtime.h>

// ---------------------------------------------------------------------------
// CDNA5 (gfx1250) point-transformer upsampling block, fp32 WMMA throughout.
// Shapes: B=4, N1=4096, N2=1024, DS=128, DD=64, DOUT=256 (derived from sizes).
// Waves compute 16x64 output strips: 1 A-fragment (b64 load) feeds 4 WMMAs.
// ---------------------------------------------------------------------------

typedef float v2f __attribute__((ext_vector_type(2)));
typedef float v8f __attribute__((ext_vector_type(8)));

#define LDH 132   // padded LDS row stride (floats) for 16x128 tiles

#define WMMA_F32(a, b, c) \
  __builtin_amdgcn_wmma_f32_16x16x4_f32(false, (a), false, (b), (short)0, (c), false, false)

// 16x64 strip of A(16xK)*W(KxN), A staged in LDS (stride LDH).
// f32 WMMA fragment layout (ISA 7.12.2):
//   A 16x4 : lane m=L&15, hi=L>>4 holds K = 2*hi + {0,1}
//   B 4x16 : lane n=L&15 holds     K = 2*hi + {0,1}
//   C 16x16: VGPR r, lane L -> row r + 8*hi, col L&15
template <int KSTEPS>
__device__ __forceinline__ void wmma_strip_lds(const float* __restrict__ hbuf,
                                               const float* __restrict__ W,
                                               int N, int ns, int lane,
                                               v8f& a0, v8f& a1, v8f& a2, v8f& a3) {
  int m = lane & 15, hi = lane >> 4;
  int col0 = ns * 64 + m;
#pragma unroll 8
  for (int kc = 0; kc < KSTEPS; ++kc) {
    int kk = kc * 4 + 2 * hi;
    v2f a = *(const v2f*)(hbuf + m * LDH + kk);          // ds_load_b64
    const float* w0 = W + (size_t)kk * N + col0;
    const float* w1 = w0 + N;
    v2f b0 = {w0[0],  w1[0]};
    v2f b1 = {w0[16], w1[16]};
    v2f b2 = {w0[32], w1[32]};
    v2f b3 = {w0[48], w1[48]};
    a0 = WMMA_F32(a, b0, a0);
    a1 = WMMA_F32(a, b1, a1);
    a2 = WMMA_F32(a, b2, a2);
    a3 = WMMA_F32(a, b3, a3);
  }
}

// ---------------------------------------------------------------------------
// Generic GEMM: C[M,N] = concat(A1[M,K1], A2[M,K2]) @ W[K1+K2, N] (+ bias)
// One wave computes one 16x64 strip; 8 waves / block. N % 64 == 0.
// ---------------------------------------------------------------------------
__global__ __launch_bounds__(256)
void gemm_wmma(const float* __restrict__ A1, int K1,
               const float* __restrict__ A2, int K2,
               const float* __restrict__ W, const float* __restrict__ bias,
               float* __restrict__ C, int M, int N) {
  int lane = threadIdx.x & 31;
  int wave = threadIdx.x >> 5;
  int strips_n = N >> 6;
  int tiles_m = M >> 4;
  int strip = blockIdx.x * (blockDim.x >> 5) + wave;
  if (strip >= tiles_m * strips_n) return;
  int mt = strip / strips_n;
  int ns = strip - mt * strips_n;

  int m = lane & 15, hi = lane >> 4;
  int arow = mt * 16 + m;
  int col0 = ns * 64 + m;
  int K = K1 + K2;

  v8f a0 = {}, a1 = {}, a2 = {}, a3 = {};
#pragma unroll 4
  for (int kk = 2 * hi; kk < K; kk += 4) {
    const float* ap = (kk < K1) ? (A1 + (size_t)arow * K1 + kk)
                                : (A2 + (size_t)arow * K2 + (kk - K1));
    v2f a = *(const v2f*)ap;                              // global_load_b64
    const float* w0 = W + (size_t)kk * N + col0;
    const float* w1 = w0 + N;
    v2f b0 = {w0[0],  w1[0]};
    v2f b1 = {w0[16], w1[16]};
    v2f b2 = {w0[32], w1[32]};
    v2f b3 = {w0[48], w1[48]};
    a0 = WMMA_F32(a, b0, a0);
    a1 = WMMA_F32(a, b1, a1);
    a2 = WMMA_F32(a, b2, a2);
    a3 = WMMA_F32(a, b3, a3);
  }
  float bc0 = bias ? bias[col0]      : 0.f;
  float bc1 = bias ? bias[col0 + 16] : 0.f;
  float bc2 = bias ? bias[col0 + 32] : 0.f;
  float bc3 = bias ? bias[col0 + 48] : 0.f;
  float* Crow = C + (size_t)(mt * 16 + 8 * hi) * N + col0;
#pragma unroll
  for (int r = 0; r < 8; ++r) {
    float* p = Crow + (size_t)r * N;
    p[0]  = a0[r] + bc0;
    p[16] = a1[r] + bc1;
    p[32] = a2[r] + bc2;
    p[48] = a3[r] + bc3;
  }
}

// ---------------------------------------------------------------------------
// Column statistics: sum / sumsq over rows (atomic partial reduction).
// blockDim.x == N, grid strides over rows.
// ---------------------------------------------------------------------------
__global__ void colstats(const float* __restrict__ X, int M, int N,
                         float* __restrict__ sum, float* __restrict__ sq) {
  int c = threadIdx.x;
  float s = 0.f, ss = 0.f;
  for (int r = blockIdx.x; r < M; r += gridDim.x) {
    float v = X[(size_t)r * N + c];
    s += v; ss += v * v;
  }
  atomicAdd(&sum[c], s);
  atomicAdd(&sq[c], ss);
}

// alpha = g*rsqrt(var+eps), beta = bt - mean*alpha  (so bn(x)=alpha*x+beta)
__global__ void finalize_stats(const float* __restrict__ sum,
                               const float* __restrict__ sq,
                               const float* __restrict__ g,
                               const float* __restrict__ bt,
                               float* __restrict__ alpha,
                               float* __restrict__ beta, float invM) {
  int c = threadIdx.x;
  float mean = sum[c] * invM;
  float var = sq[c] * invM - mean * mean;
  float a = g[c] * rsqrtf(var + 1e-5f);
  alpha[c] = a;
  beta[c] = bt[c] - mean * a;
}

__global__ void bn_relu(const float* __restrict__ X,
                        const float* __restrict__ alpha,
                        const float* __restrict__ beta,
                        float* __restrict__ Y, int total, int N) {
  int t = blockIdx.x * blockDim.x + threadIdx.x;
  if (t < total) {
    int c = t % N;
    Y[t] = fmaxf(alpha[c] * X[t] + beta[c], 0.f);
  }
}

// ---------------------------------------------------------------------------
// kNN (k=8) of p1 against p2 within each batch; stores global row ids and
// normalized inverse-distance weights. One thread per query point, candidates
// staged through LDS; top-8 kept sorted (by squared distance) in registers.
// ---------------------------------------------------------------------------
__global__ __launch_bounds__(256)
void knn8_interp(const float* __restrict__ p1, const float* __restrict__ p2,
                 int N1, int N2, int* __restrict__ iidx, float* __restrict__ iw) {
  __shared__ float tx[256], ty[256], tz[256];
  int i = blockIdx.x * 256 + threadIdx.x;
  int b = i / N1;                      // blocks never straddle batches
  float px = p1[i * 3], py = p1[i * 3 + 1], pz = p1[i * 3 + 2];
  float d8[8]; int i8[8];
#pragma unroll
  for (int s = 0; s < 8; ++s) { d8[s] = 3.4e38f; i8[s] = 0; }

  for (int t0 = 0; t0 < N2; t0 += 256) {
    int cj = (b * N2 + t0 + threadIdx.x) * 3;
    tx[threadIdx.x] = p2[cj]; ty[threadIdx.x] = p2[cj + 1]; tz[threadIdx.x] = p2[cj + 2];
    __syncthreads();
    for (int j = 0; j < 256; ++j) {
      float dx = px - tx[j], dy = py - ty[j], dz = pz - tz[j];
      float dd = dx * dx + dy * dy + dz * dz;
      if (dd < d8[7]) {
        d8[7] = dd; i8[7] = t0 + j;
#pragma unroll
        for (int s = 7; s > 0; --s) {
          if (d8[s] < d8[s - 1]) {
            float td = d8[s]; d8[s] = d8[s - 1]; d8[s - 1] = td;
            int ti = i8[s]; i8[s] = i8[s - 1]; i8[s - 1] = ti;
          }
        }
      }
    }
    __syncthreads();
  }
  float r[8], tot = 0.f;
#pragma unroll
  for (int s = 0; s < 8; ++s) { r[s] = 1.0f / (sqrtf(d8[s]) + 1e-8f); tot += r[s]; }
#pragma unroll
  for (int s = 0; s < 8; ++s) {
    iw[i * 8 + s] = r[s] / tot;
    iidx[i * 8 + s] = b * N2 + i8[s];
  }
}

// Inverse-distance weighted feature gather: x2i[i] = sum_j w_j * x2[idx_j]
__global__ void interp_gather(const float* __restrict__ x2,
                              const int* __restrict__ iidx,
                              const float* __restrict__ iw,
                              float* __restrict__ x2i, int DS) {
  int t = blockIdx.x * blockDim.x + threadIdx.x;
  int chunks = DS >> 2;
  int i = t / chunks;
  int c = (t - i * chunks) << 2;
  const float4* xv = (const float4*)x2;
  float4 acc = {0.f, 0.f, 0.f, 0.f};
#pragma unroll
  for (int j = 0; j < 8; ++j) {
    int gr = iidx[i * 8 + j];
    float w = iw[i * 8 + j];
    float4 f = xv[((size_t)gr * DS + c) >> 2];
    acc.x += w * f.x; acc.y += w * f.y; acc.z += w * f.z; acc.w += w * f.w;
  }
  ((float4*)x2i)[t] = acc;
}

// kNN (k=16) of p1 against p1 (self included); stores global row ids.
__global__ __launch_bounds__(256)
void knn16_self(const float* __restrict__ p1, int N1, int* __restrict__ nidx) {
  __shared__ float tx[256], ty[256], tz[256];
  int i = blockIdx.x * 256 + threadIdx.x;
  int b = i / N1;
  float px = p1[i * 3], py = p1[i * 3 + 1], pz = p1[i * 3 + 2];
  float dk[16]; int ik[16];
#pragma unroll
  for (int s = 0; s < 16; ++s) { dk[s] = 3.4e38f; ik[s] = 0; }

  for (int t0 = 0; t0 < N1; t0 += 256) {
    int cj = (b * N1 + t0 + threadIdx.x) * 3;
    tx[threadIdx.x] = p1[cj]; ty[threadIdx.x] = p1[cj + 1]; tz[threadIdx.x] = p1[cj + 2];
    __syncthreads();
    for (int j = 0; j < 256; ++j) {
      float dx = px - tx[j], dy = py - ty[j], dz = pz - tz[j];
      float dd = dx * dx + dy * dy + dz * dz;
      if (dd < dk[15]) {
        dk[15] = dd; ik[15] = t0 + j;
#pragma unroll
        for (int s = 15; s > 0; --s) {
          if (dk[s] < dk[s - 1]) {
            float td = dk[s]; dk[s] = dk[s - 1]; dk[s - 1] = td;
            int ti = ik[s]; ik[s] = ik[s - 1]; ik[s - 1] = ti;
          }
        }
      }
    }
    __syncthreads();
  }
#pragma unroll
  for (int s = 0; s < 16; ++s) nidx[i * 16 + s] = b * N1 + ik[s];
}

// ---------------------------------------------------------------------------
// Attention pass A: per point build h = q - k[g] + xe[g] (16x128), run
// H1 = h@W1 + b1 via WMMA strips, accumulate per-column sum/sumsq for the
// BN batch statistics (262144 rows). One wave per point.
// ---------------------------------------------------------------------------
__global__ __launch_bounds__(32)
void attn_pass_a(const float* __restrict__ q, const float* __restrict__ k,
                 const float* __restrict__ xe, const int* __restrict__ nidx,
                 const float* __restrict__ W1, const float* __restrict__ b1,
                 float* __restrict__ csum, float* __restrict__ csq) {
  __shared__ float h[16 * LDH];
  __shared__ int g[16];
  int i = blockIdx.x, lane = threadIdx.x;
  if (lane < 16) g[lane] = nidx[i * 16 + lane];
  __syncthreads();
  for (int t = lane; t < 16 * 128; t += 32) {
    int m = t >> 7, c = t & 127;
    int gr = g[m];
    h[m * LDH + c] = q[(size_t)i * 128 + c] - k[(size_t)gr * 128 + c]
                   + xe[(size_t)gr * 128 + c];
  }
  __syncthreads();
  int m = lane & 15, hi = lane >> 4;

  auto emit = [&](const v8f& acc, int col) {
    float bb = b1[col];
    float s = 0.f, ss = 0.f;
#pragma unroll
    for (int r = 0; r < 8; ++r) { float y = acc[r] + bb; s += y; ss += y * y; }
    s += __shfl_xor(s, 16);
    ss += __shfl_xor(ss, 16);
    if (hi == 0) { atomicAdd(&csum[col], s); atomicAdd(&csq[col], ss); }
  };

  for (int ns = 0; ns < 2; ++ns) {
    v8f a0 = {}, a1 = {}, a2 = {}, a3 = {};
    wmma_strip_lds<32>(h, W1, 128, ns, lane, a0, a1, a2, a3);
    int col0 = ns * 64 + m;
    emit(a0, col0); emit(a1, col0 + 16); emit(a2, col0 + 32); emit(a3, col0 + 48);
  }
}

// ---------------------------------------------------------------------------
// Attention pass B: recompute h, H2 = relu(bn(h@W1+b1)), sim = H2@W2+b2,
// softmax over the 16 neighbors per channel, aggregate with v2 = v[g]+xe[g].
// One wave per point; two 16x128x128 WMMA GEMMs fully in LDS.
// ---------------------------------------------------------------------------
__global__ __launch_bounds__(32)
void attn_pass_b(const float* __restrict__ q, const float* __restrict__ k,
                 const float* __restrict__ v, const float* __restrict__ xe,
                 const int* __restrict__ nidx,
                 const float* __restrict__ W1, const float* __restrict__ b1,
                 const float* __restrict__ alpha, const float* __restrict__ beta,
                 const float* __restrict__ W2, const float* __restrict__ b2,
                 float* __restrict__ agg) {
  __shared__ float h[16 * LDH];
  __shared__ float h2[16 * LDH];
  __shared__ float v2b[16 * LDH];
  __shared__ int g[16];
  int i = blockIdx.x, lane = threadIdx.x;
  if (lane < 16) g[lane] = nidx[i * 16 + lane];
  __syncthreads();
  for (int t = lane; t < 16 * 128; t += 32) {
    int m = t >> 7, c = t & 127;
    int gr = g[m];
    float xv = xe[(size_t)gr * 128 + c];
    h[m * LDH + c] = q[(size_t)i * 128 + c] - k[(size_t)gr * 128 + c] + xv;
    v2b[m * LDH + c] = v[(size_t)gr * 128 + c] + xv;
  }
  __syncthreads();
  int m = lane & 15, hi = lane >> 4;

  // H2 = relu(alpha*(h@W1+b1)+beta)
  {
    auto emit = [&](const v8f& acc, int col) {
      float bb = b1[col], al = alpha[col], be = beta[col];
#pragma unroll
      for (int r = 0; r < 8; ++r) {
        float y = al * (acc[r] + bb) + be;
        h2[(r + 8 * hi) * LDH + col] = fmaxf(y, 0.f);
      }
    };
    for (int ns = 0; ns < 2; ++ns) {
      v8f a0 = {}, a1 = {}, a2 = {}, a3 = {};
      wmma_strip_lds<32>(h, W1, 128, ns, lane, a0, a1, a2, a3);
      int col0 = ns * 64 + m;
      emit(a0, col0); emit(a1, col0 + 16); emit(a2, col0 + 32); emit(a3, col0 + 48);
    }
  }
  __syncthreads();
  // sim = H2@W2 + b2  (reuse h)
  {
    auto emit = [&](const v8f& acc, int col) {
      float bb = b2[col];
#pragma unroll
      for (int r = 0; r < 8; ++r)
        h[(r + 8 * hi) * LDH + col] = acc[r] + bb;
    };
    for (int ns = 0; ns < 2; ++ns) {
      v8f a0 = {}, a1 = {}, a2 = {}, a3 = {};
      wmma_strip_lds<32>(h2, W2, 128, ns, lane, a0, a1, a2, a3);
      int col0 = ns * 64 + m;
      emit(a0, col0); emit(a1, col0 + 16); emit(a2, col0 + 32); emit(a3, col0 + 48);
    }
  }
  __syncthreads();
  // softmax over 16 neighbors per channel + weighted aggregation
  for (int c = lane; c < 128; c += 32) {
    float mx = -3.4e38f;
#pragma unroll
    for (int mm = 0; mm < 16; ++mm) mx = fmaxf(mx, h[mm * LDH + c]);
    float se = 0.f, out = 0.f;
#pragma unroll
    for (int mm = 0; mm < 16; ++mm) {
      float e = __expf(h[mm * LDH + c] - mx);
      se += e;
      out += e * v2b[mm * LDH + c];
    }
    agg[(size_t)i * 128 + c] = out / se;
  }
}

// ---------------------------------------------------------------------------
extern "C" void kernel_launch(void* const* d_in, const int* in_sizes, int n_in,
                              void* d_out, int out_size, void* d_ws, size_t ws_size,
                              hipStream_t stream) {
  (void)n_in; (void)out_size; (void)ws_size;
  const float* p1  = (const float*)d_in[0];
  const float* x1  = (const float*)d_in[1];
  const int*   o1  = (const int*)d_in[2];
  const float* p2  = (const float*)d_in[3];
  const float* x2  = (const float*)d_in[4];
  const float* Wd  = (const float*)d_in[7];
  const float* bd  = (const float*)d_in[8];
  const float* gd  = (const float*)d_in[9];
  const float* btd = (const float*)d_in[10];
  const float* Wq  = (const float*)d_in[11];
  const float* Wk  = (const float*)d_in[12];
  const float* Wv  = (const float*)d_in[13];
  const float* W1  = (const float*)d_in[14];
  const float* b1  = (const float*)d_in[15];
  const float* g1  = (const float*)d_in[16];
  const float* bt1 = (const float*)d_in[17];
  const float* W2  = (const float*)d_in[18];
  const float* b2  = (const float*)d_in[19];
  const float* Wo  = (const float*)d_in[20];
  const float* bo  = (const float*)d_in[21];
  const float* go  = (const float*)d_in[22];
  const float* bto = (const float*)d_in[23];

  const int B    = in_sizes[2];
  const int N1   = in_sizes[0] / (3 * B);
  const int N2   = in_sizes[3] / (3 * B);
  const int M1   = B * N1;                 // 16384
  const int M2   = B * N2;                 // 4096
  const int DD   = in_sizes[1] / M1;       // 64
  const int DS   = in_sizes[4] / M2;       // 128
  const int DOUT = in_sizes[21];           // 256

  // workspace layout (floats)
  float* ws = (float*)d_ws;
  size_t o = 0;
  float* x1d  = ws + o; o += (size_t)M1 * DS;
  float* x2i  = ws + o; o += (size_t)M1 * DS;
  float* qb   = ws + o; o += (size_t)M1 * DS;
  float* kb   = ws + o; o += (size_t)M1 * DS;
  float* vb   = ws + o; o += (size_t)M1 * DS;
  float* aggb = ws + o; o += (size_t)M1 * DS;
  float* tmp  = ws + o; o += (size_t)M1 * DOUT;
  int*   nidx = (int*)(ws + o); o += (size_t)M1 * 16;
  int*   iidx = (int*)(ws + o); o += (size_t)M1 * 8;
  float* iw   = ws + o; o += (size_t)M1 * 8;
  float* S    = ws + o;                           // stats: 2048 floats
  float *Ssum = S, *Ssq = S + 256, *Sal = S + 512, *Sbe = S + 768;
  float* S2   = S + 1024;
  float *S2sum = S2, *S2sq = S2 + 128, *S2al = S2 + 256, *S2be = S2 + 384;

  hipMemsetAsync(S, 0, 2048 * sizeof(float), stream);

  // 1. dense MLP: x1 @ W_dense + b, BN(batch stats) + ReLU -> x1d
  {
    int strips = (M1 / 16) * (DS / 64);
    gemm_wmma<<<dim3((strips + 7) / 8), dim3(256), 0, stream>>>(
        x1, DD, nullptr, 0, Wd, bd, tmp, M1, DS);
  }
  colstats<<<dim3(64), dim3(DS), 0, stream>>>(tmp, M1, DS, Ssum, Ssq);
  finalize_stats<<<dim3(1), dim3(DS), 0, stream>>>(Ssum, Ssq, gd, btd, Sal, Sbe,
                                                   1.0f / (float)M1);
  bn_relu<<<dim3((M1 * DS) / 256), dim3(256), 0, stream>>>(tmp, Sal, Sbe, x1d,
                                                           M1 * DS, DS);

  // 2. inverse-distance kNN interpolation (k=8)
  knn8_interp<<<dim3(M1 / 256), dim3(256), 0, stream>>>(p1, p2, N1, N2, iidx, iw);
  interp_gather<<<dim3((M1 * DS / 4) / 256), dim3(256), 0, stream>>>(x2, iidx, iw,
                                                                     x2i, DS);

  // 3. 16-NN on p1 within batch
  knn16_self<<<dim3(M1 / 256), dim3(256), 0, stream>>>(p1, N1, nidx);

  // 4. q/k/v projections
  {
    int strips = (M1 / 16) * (DS / 64);
    dim3 gr((strips + 7) / 8);
    gemm_wmma<<<gr, dim3(256), 0, stream>>>(x1d, DS, nullptr, 0, Wq, nullptr, qb, M1, DS);
    gemm_wmma<<<gr, dim3(256), 0, stream>>>(x2i, DS, nullptr, 0, Wk, nullptr, kb, M1, DS);
    gemm_wmma<<<gr, dim3(256), 0, stream>>>(x2i, DS, nullptr, 0, Wv, nullptr, vb, M1, DS);
  }

  // 5. attention: stats pass, BN finalize, fused pass B
  attn_pass_a<<<dim3(M1), dim3(32), 0, stream>>>(qb, kb, x1d, nidx, W1, b1,
                                                 S2sum, S2sq);
  finalize_stats<<<dim3(1), dim3(DS), 0, stream>>>(S2sum, S2sq, g1, bt1, S2al, S2be,
                                                   1.0f / (float)(M1 * 16));
  attn_pass_b<<<dim3(M1), dim3(32), 0, stream>>>(qb, kb, vb, x1d, nidx, W1, b1,
                                                 S2al, S2be, W2, b2, aggb);

  // 6. output MLP on concat(agg, x1): GEMM + BN + ReLU -> d_out (x part)
  hipMemsetAsync(S, 0, 1024 * sizeof(float), stream);
  {
    int strips = (M1 / 16) * (DOUT / 64);
    gemm_wmma<<<dim3((strips + 7) / 8), dim3(256), 0, stream>>>(
        aggb, DS, x1, DD, Wo, bo, tmp, M1, DOUT);
  }
  colstats<<<dim3(64), dim3(DOUT), 0, stream>>>(tmp, M1, DOUT, Ssum, Ssq);
  finalize_stats<<<dim3(1), dim3(DOUT), 0, stream>>>(Ssum, Ssq, go, bto, Sal, Sbe,
                                                     1.0f / (float)M1);
  float* outx = (float*)d_out + (size_t)M1 * 3;
  bn_relu<<<dim3((M1 * DOUT) / 256), dim3(256), 0, stream>>>(tmp, Sal, Sbe, outx,
                                                             M1 * DOUT, DOUT);

  // 7. tuple passthroughs: p1 (front) and o1 (tail, int bits)
  hipMemcpyAsync(d_out, p1, (size_t)M1 * 3 * sizeof(float),
                 hipMemcpyDeviceToDevice, stream);
  hipMemcpyAsync((char*)d_out + ((size_t)M1 * 3 + (size_t)M1 * DOUT) * sizeof(float),
                 o1, (size_t)B * sizeof(int), hipMemcpyDeviceToDevice, stream);
}